// TransformerClassifPerProtECHierarchy_89051851915584
// MI455X (gfx1250) — compile-verified
//
#include <hip/hip_runtime.h>

#define D_MODEL     1024
#define NUM_CLASSES 5242
#define NC_PAD      5248        // 41 * 128
#define PER_LVL     256
#define B_ROWS      8192
#define LN_EPS      1e-5f

#define MT 128
#define NT 128
#define KC 32
#define LDSS 36                 // LDS row stride (dwords): conflict-free, 16B aligned

typedef __attribute__((ext_vector_type(2))) float v2f;
typedef __attribute__((ext_vector_type(8))) float v8f;
typedef __attribute__((ext_vector_type(4))) int   v4i;

#define AS1 __attribute__((address_space(1)))
#define AS3 __attribute__((address_space(3)))

#if defined(__gfx1250__) && __has_builtin(__builtin_amdgcn_global_load_async_to_lds_b128)
#define HAVE_ASYNC_LDS 1
#else
#define HAVE_ASYNC_LDS 0
#endif

static __device__ __forceinline__ void wait_async0() {
#if __has_builtin(__builtin_amdgcn_s_wait_asynccnt)
  __builtin_amdgcn_s_wait_asynccnt(0);
#else
  asm volatile("s_wait_asynccnt 0x0" ::: "memory");
#endif
}

// ---------------------------------------------------------------------------
// Kernel 1: gather the hierarchical embedding rows into Wt[NC_PAD][1024]
// (class-major; pad rows zeroed so the GEMM needs no load guards)
// ---------------------------------------------------------------------------
__global__ __launch_bounds__(256) void gather_weight_kernel(
    const float* __restrict__ W1, const float* __restrict__ W2,
    const float* __restrict__ W3, const float* __restrict__ W4,
    const int* __restrict__ idx1, const int* __restrict__ idx2,
    const int* __restrict__ idx3, const int* __restrict__ idx4,
    float* __restrict__ Wt) {
  const int c = blockIdx.x;
  const int t = threadIdx.x;
  float* row = Wt + (size_t)c * D_MODEL;
  if (c < NUM_CLASSES) {
    row[0 * PER_LVL + t] = W1[idx1[c] * PER_LVL + t];
    row[1 * PER_LVL + t] = W2[idx2[c] * PER_LVL + t];
    row[2 * PER_LVL + t] = W3[idx3[c] * PER_LVL + t];
    row[3 * PER_LVL + t] = W4[idx4[c] * PER_LVL + t];
  } else {
    row[0 * PER_LVL + t] = 0.0f;
    row[1 * PER_LVL + t] = 0.0f;
    row[2 * PER_LVL + t] = 0.0f;
    row[3 * PER_LVL + t] = 0.0f;
  }
}

// ---------------------------------------------------------------------------
// Kernel 2: Out[8192, 5242] = X[8192,1024] @ Wt^T + bias, via
// V_WMMA_F32_16X16X4_F32. 8 waves/block, 128x128 tile, waves 4(M) x 2(N),
// each wave: 32x64 patch = 2x4 accumulators of 16x16. Panels staged in LDS
// via GLOBAL_LOAD_ASYNC_TO_LDS_B128 (ASYNCcnt) when available.
// ---------------------------------------------------------------------------
__global__ __launch_bounds__(256) void gemm_bias_kernel(
    const float* __restrict__ X, const float* __restrict__ Wt,
    const float* __restrict__ bias, float* __restrict__ Out) {
  __shared__ float As[MT][LDSS];   // X tile:  rows = M, cols = K chunk
  __shared__ float Bs[NT][LDSS];   // Wt tile: rows = N(class), cols = K chunk

  const int tid  = threadIdx.x;
  const int lane = tid & 31;
  const int wave = tid >> 5;
  const int wm   = wave & 3;       // wave M offset: wm*32
  const int wn   = wave >> 2;      // wave N offset: wn*64
  const int m0   = blockIdx.y * MT;
  const int n0   = blockIdx.x * NT;

  // Fragment addressing per ISA 7.12.2 (32-bit A 16x4 / B 4x16):
  // lanes 0-15 hold K={0,1}, lanes 16-31 hold K={2,3}, row = lane % 16.
  const int lrow = lane & 15;
  const int lk   = (lane >> 4) * 2;

  v8f acc[2][4];
#pragma unroll
  for (int mi = 0; mi < 2; ++mi)
#pragma unroll
    for (int ni = 0; ni < 4; ++ni)
      acc[mi][ni] = (v8f){0.f, 0.f, 0.f, 0.f, 0.f, 0.f, 0.f, 0.f};

  for (int kc = 0; kc < D_MODEL; kc += KC) {
    // Stage A and B panels: 128 rows x 32 cols each = 1024 float4 per panel,
    // 256 threads -> 4 x b128 per thread per panel.
#if HAVE_ASYNC_LDS
#pragma unroll
    for (int i = 0; i < 4; ++i) {
      const int f  = tid + i * 256;
      const int r  = f >> 3;
      const int c4 = (f & 7) * 4;
      __builtin_amdgcn_global_load_async_to_lds_b128(
          (AS1 v4i*)(uintptr_t)&X[(size_t)(m0 + r) * D_MODEL + kc + c4],
          (AS3 v4i*)&As[r][c4], 0, 0);
      __builtin_amdgcn_global_load_async_to_lds_b128(
          (AS1 v4i*)(uintptr_t)&Wt[(size_t)(n0 + r) * D_MODEL + kc + c4],
          (AS3 v4i*)&Bs[r][c4], 0, 0);
    }
    wait_async0();
#else
#pragma unroll
    for (int i = 0; i < 4; ++i) {
      const int f  = tid + i * 256;
      const int r  = f >> 3;
      const int c4 = (f & 7) * 4;
      *(float4*)&As[r][c4] =
          *(const float4*)&X[(size_t)(m0 + r) * D_MODEL + kc + c4];
      *(float4*)&Bs[r][c4] =
          *(const float4*)&Wt[(size_t)(n0 + r) * D_MODEL + kc + c4];
    }
    // Prefetch next K panels into cache (global_prefetch_b8).
    if (kc + KC < D_MODEL) {
      const int pr = tid & 127;
      __builtin_prefetch(&X[(size_t)(m0 + pr) * D_MODEL + kc + KC], 0, 3);
      __builtin_prefetch(&Wt[(size_t)(n0 + pr) * D_MODEL + kc + KC], 0, 3);
    }
#endif
    __syncthreads();

#pragma unroll
    for (int ks = 0; ks < KC; ks += 4) {
      v2f a[2], b[4];
#pragma unroll
      for (int mi = 0; mi < 2; ++mi)
        a[mi] = *(const v2f*)&As[wm * 32 + mi * 16 + lrow][ks + lk];
#pragma unroll
      for (int ni = 0; ni < 4; ++ni)
        b[ni] = *(const v2f*)&Bs[wn * 64 + ni * 16 + lrow][ks + lk];
#pragma unroll
      for (int mi = 0; mi < 2; ++mi)
#pragma unroll
        for (int ni = 0; ni < 4; ++ni)
          acc[mi][ni] = __builtin_amdgcn_wmma_f32_16x16x4_f32(
              /*neg_a=*/false, a[mi], /*neg_b=*/false, b[ni],
              /*c_mod=*/(short)0, acc[mi][ni],
              /*reuse_a=*/false, /*reuse_b=*/false);
    }
    __syncthreads();
  }

  // Store + fused bias. C/D layout: VGPR r -> M = r (lanes 0-15) / r+8
  // (lanes 16-31), N = lane % 16.
  const int nbase = n0 + wn * 64;
  const int mbase = m0 + wm * 32 + ((lane >> 4) * 8);
#pragma unroll
  for (int mi = 0; mi < 2; ++mi) {
#pragma unroll
    for (int ni = 0; ni < 4; ++ni) {
      const int col = nbase + ni * 16 + lrow;
      if (col < NUM_CLASSES) {
        const float bv = bias[col];
#pragma unroll
        for (int r = 0; r < 8; ++r) {
          const int row = mbase + mi * 16 + r;
          Out[(size_t)row * NUM_CLASSES + col] = acc[mi][ni][r] + bv;
        }
      }
    }
  }
}

// ---------------------------------------------------------------------------
// Kernel 3: in-place LayerNorm over the class axis (5242), affine.
// One block per batch row; values cached in registers between passes.
// ---------------------------------------------------------------------------
__global__ __launch_bounds__(256) void layernorm_kernel(
    float* __restrict__ Out, const float* __restrict__ gamma,
    const float* __restrict__ beta) {
  __shared__ float red[256];
  const int t = threadIdx.x;
  float* p = Out + (size_t)blockIdx.x * NUM_CLASSES;

  float vals[(NUM_CLASSES + 255) / 256];
  float s = 0.f, sq = 0.f;
  int n = 0;
  for (int c = t; c < NUM_CLASSES; c += 256, ++n) {
    const float v = p[c];
    vals[n] = v;
    s += v;
    sq += v * v;
  }

  red[t] = s;
  __syncthreads();
  for (int off = 128; off > 0; off >>= 1) {
    if (t < off) red[t] += red[t + off];
    __syncthreads();
  }
  const float mean = red[0] * (1.0f / NUM_CLASSES);
  __syncthreads();

  red[t] = sq;
  __syncthreads();
  for (int off = 128; off > 0; off >>= 1) {
    if (t < off) red[t] += red[t + off];
    __syncthreads();
  }
  const float var  = red[0] * (1.0f / NUM_CLASSES) - mean * mean;
  const float rstd = rsqrtf(var + LN_EPS);

  n = 0;
  for (int c = t; c < NUM_CLASSES; c += 256, ++n)
    p[c] = (vals[n] - mean) * rstd * gamma[c] + beta[c];
}

// ---------------------------------------------------------------------------
extern "C" void kernel_launch(void* const* d_in, const int* in_sizes, int n_in,
                              void* d_out, int out_size, void* d_ws,
                              size_t ws_size, hipStream_t stream) {
  const float* x     = (const float*)d_in[0];
  const float* W1    = (const float*)d_in[1];
  const float* W2    = (const float*)d_in[2];
  const float* W3    = (const float*)d_in[3];
  const float* W4    = (const float*)d_in[4];
  const float* bias  = (const float*)d_in[5];
  const float* gamma = (const float*)d_in[6];
  const float* beta  = (const float*)d_in[7];
  const int*   idx1  = (const int*)d_in[8];
  const int*   idx2  = (const int*)d_in[9];
  const int*   idx3  = (const int*)d_in[10];
  const int*   idx4  = (const int*)d_in[11];

  float* out = (float*)d_out;
  float* Wt  = (float*)d_ws;  // NC_PAD * 1024 floats = 21.5 MB

  gather_weight_kernel<<<NC_PAD, 256, 0, stream>>>(W1, W2, W3, W4, idx1, idx2,
                                                   idx3, idx4, Wt);

  dim3 grid(NC_PAD / NT, B_ROWS / MT);
  gemm_bias_kernel<<<grid, 256, 0, stream>>>(x, Wt, bias, out);

  layernorm_kernel<<<B_ROWS, 256, 0, stream>>>(out, gamma, beta);
}